// PathLossTwoGaussWeighted_34763465294043
// MI455X (gfx1250) — compile-verified
//
#include <hip/hip_runtime.h>
#include <hip/hip_bf16.h>
#include <math.h>

typedef __attribute__((ext_vector_type(2))) float v2f;
typedef __attribute__((ext_vector_type(8))) float v8f;

#define BIGC 1000000.0f
#define EPSC 1e-8f

// Problem dims (fixed by reference)
#define NB   32
#define NP   512
#define NG   512

// Workspace layout (offsets in floats)
#define OFF_SEGA   0         // float4[NB*NG]  seg (unscaled, w=0)
#define OFF_SEGB   65536     // float4[NB*NG]  seg/sigma^2
#define OFF_SEGC   131072    // float4[NB*NG]  p0/sigma^2
#define OFF_TGTS   196608    // float4[NB*NG]  tgt/sigma
#define OFF_SCAL   262144    // float4[NB*NG]  {p0.seg, 1/len2, p0^2sc(+BIG), p0.seg sc}
#define OFF_SCAL2  327680    // float4[NB*NG]  {seg^2sc, tgt^2sc(+BIG), 0, 0}
#define OFF_SIGI   393216    // float4[NB]     1/sigma
#define OFF_SIGI2  393344    // float4[NB]     1/sigma^2
#define OFF_PATHW  393472    // float[1024]    per-wave sum (1-label)*pm
#define OFF_PATHM  394496    // float[1024]    per-wave sum pm
#define OFF_SMOOTH 395520    // float[NB]
#define OFF_CAE    395552    // float[256]

// -------- Kernel 1: per-batch extent -> sigma (shared for line & point) -----
__global__ __launch_bounds__(256) void k_sigma(const float* __restrict__ pp,
                                               float4* __restrict__ sigInv,
                                               float4* __restrict__ sigInv2) {
  int b = blockIdx.x;
  int tid = threadIdx.x;
  const float* base = pp + (size_t)b * NP * 3;
  float mx0 = -3.0e38f, mx1 = -3.0e38f, mx2 = -3.0e38f;
  float mn0 =  3.0e38f, mn1 =  3.0e38f, mn2 =  3.0e38f;
  for (int i = tid; i < NP; i += 256) {
    float x = base[i * 3 + 0], y = base[i * 3 + 1], z = base[i * 3 + 2];
    mx0 = fmaxf(mx0, x); mx1 = fmaxf(mx1, y); mx2 = fmaxf(mx2, z);
    mn0 = fminf(mn0, x); mn1 = fminf(mn1, y); mn2 = fminf(mn2, z);
  }
  __shared__ float s0[256], s1[256], s2[256], s3[256], s4[256], s5[256];
  s0[tid] = mx0; s1[tid] = mx1; s2[tid] = mx2;
  s3[tid] = mn0; s4[tid] = mn1; s5[tid] = mn2;
  __syncthreads();
  for (int off = 128; off > 0; off >>= 1) {
    if (tid < off) {
      s0[tid] = fmaxf(s0[tid], s0[tid + off]);
      s1[tid] = fmaxf(s1[tid], s1[tid + off]);
      s2[tid] = fmaxf(s2[tid], s2[tid + off]);
      s3[tid] = fminf(s3[tid], s3[tid + off]);
      s4[tid] = fminf(s4[tid], s4[tid + off]);
      s5[tid] = fminf(s5[tid], s5[tid + off]);
    }
    __syncthreads();
  }
  if (tid == 0) {
    float sx = fmaxf(0.05f * (s0[0] - s3[0]), EPSC);
    float sy = fmaxf(0.05f * (s1[0] - s4[0]), EPSC);
    float sz = fmaxf(0.05f * (s2[0] - s5[0]), EPSC);
    sigInv[b]  = make_float4(1.0f / sx, 1.0f / sy, 1.0f / sz, 0.0f);
    sigInv2[b] = make_float4(1.0f / (sx * sx), 1.0f / (sy * sy), 1.0f / (sz * sz), 0.0f);
  }
}

// -------- Kernel 2: per-(b,m) precompute of B-matrices and column scalars ---
__global__ __launch_bounds__(256) void k_prep(const float* __restrict__ tgt,
                                              const int* __restrict__ tmask,
                                              const float4* __restrict__ sigInv,
                                              const float4* __restrict__ sigInv2,
                                              float4* __restrict__ segA,
                                              float4* __restrict__ segB,
                                              float4* __restrict__ segC,
                                              float4* __restrict__ tgtS,
                                              float4* __restrict__ scal,
                                              float4* __restrict__ scal2) {
  int idx = blockIdx.x * 256 + threadIdx.x;   // 0 .. NB*NG-1
  int b = idx >> 9;
  int m = idx & (NG - 1);
  const float* tb = tgt + ((size_t)b * NG + m) * 3;
  float t0 = tb[0], t1 = tb[1], t2v = tb[2];
  int tmk = tmask[b * NG + m];
  float4 si  = sigInv[b];
  float4 si2 = sigInv2[b];

  float4 ts = make_float4(t0 * si.x, t1 * si.y, t2v * si.z, 0.0f);
  tgtS[idx] = ts;
  float tgt2sc = ts.x * ts.x + ts.y * ts.y + ts.z * ts.z + (tmk ? 0.0f : BIGC);

  float4 sA = make_float4(0, 0, 0, 0), sB = make_float4(0, 0, 0, 0), sC = make_float4(0, 0, 0, 0);
  float p0seg = 0.0f, invl2 = 0.0f, p02 = 2.0f * BIGC, p0segsc = 0.0f, seg2sc = 0.0f;
  if (m < NG - 1) {
    float q0 = tb[3], q1 = tb[4], q2 = tb[5];
    float sx = q0 - t0, sy = q1 - t1, sz = q2 - t2v;
    float len2 = fmaxf(sx * sx + sy * sy + sz * sz, EPSC);
    int smk = tmk && tmask[b * NG + m + 1];
    sA = make_float4(sx, sy, sz, 0.0f);
    sB = make_float4(sx * si2.x, sy * si2.y, sz * si2.z, 0.0f);
    sC = make_float4(t0 * si2.x, t1 * si2.y, t2v * si2.z, 0.0f);
    p0seg  = t0 * sx + t1 * sy + t2v * sz;
    invl2  = 1.0f / len2;
    p02    = t0 * t0 * si2.x + t1 * t1 * si2.y + t2v * t2v * si2.z + (smk ? 0.0f : BIGC);
    p0segsc = t0 * sx * si2.x + t1 * sy * si2.y + t2v * sz * si2.z;
    seg2sc  = sx * sx * si2.x + sy * sy * si2.y + sz * sz * si2.z;
  }
  segA[idx] = sA; segB[idx] = sB; segC[idx] = sC;
  scal[idx]  = make_float4(p0seg, invl2, p02, p0segsc);
  scal2[idx] = make_float4(seg2sc, tgt2sc, 0.0f, 0.0f);
}

// -------- WMMA helper: D = A(16x4,f32) x B(4x16,f32) ------------------------
__device__ inline v8f wmma4(v2f a, v2f b) {
  v8f c = {};
  return __builtin_amdgcn_wmma_f32_16x16x4_f32(false, a, false, b, (short)0, c, false, false);
}

// -------- Kernel 3: fused pairwise line+point label via WMMA ----------------
// One wave = 16 path rows; 8 waves/block; grid = NB * (NP/128) = 128 blocks.
__global__ __launch_bounds__(256) void k_main(const float* __restrict__ pp,
                                              const int* __restrict__ pmask,
                                              const float4* __restrict__ segA,
                                              const float4* __restrict__ segB,
                                              const float4* __restrict__ segC,
                                              const float4* __restrict__ tgtS,
                                              const float4* __restrict__ scal,
                                              const float4* __restrict__ scal2,
                                              const float4* __restrict__ sigInv,
                                              float* __restrict__ pathW,
                                              float* __restrict__ pathM) {
  int b    = blockIdx.x >> 2;
  int tb   = blockIdx.x & 3;
  int wave = threadIdx.x >> 5;
  int lane = threadIdx.x & 31;
  int r    = lane & 15;
  int hi   = lane >> 4;
  int rowBase = tb * 128 + wave * 16;
  int row = rowBase + r;

  const float* pb = pp + ((size_t)b * NP + row) * 3;
  float px = pb[0], py = pb[1], pz = pb[2];
  float pmrow = pmask[b * NP + row] ? 1.0f : 0.0f;
  px *= pmrow; py *= pmrow; pz *= pmrow;     // pc = path_points * mask
  float4 si = sigInv[b];

  // A matrices: ISA 16x4 layout — lanes 0-15 hold K=0,1; lanes 16-31 hold K=2,3
  v2f a_ln; a_ln.x = hi ? pz : px;           a_ln.y = hi ? 0.0f : py;
  v2f a_pt; a_pt.x = hi ? pz * si.z : px * si.x; a_pt.y = hi ? 0.0f : py * si.y;

  // |pc/sigma|^2 per row (pair-sum across the two half-lanes holding the row)
  float part = a_pt.x * a_pt.x + a_pt.y * a_pt.y;
  float full = part + __shfl_xor(part, 16);
  float row2[8];
#pragma unroll
  for (int j = 0; j < 8; j++) row2[j] = __shfl(full, j + hi * 8);

  float minL[8], minP[8];
#pragma unroll
  for (int j = 0; j < 8; j++) { minL[j] = 3.0e38f; minP[j] = 3.0e38f; }

  const float4* gA  = segA  + (size_t)b * NG;
  const float4* gB  = segB  + (size_t)b * NG;
  const float4* gC  = segC  + (size_t)b * NG;
  const float4* gT  = tgtS  + (size_t)b * NG;
  const float4* gS  = scal  + (size_t)b * NG;
  const float4* gS2 = scal2 + (size_t)b * NG;

  for (int t = 0; t < NG / 16; t++) {
    int mc = t * 16 + r;
    float4 A4 = gA[mc], B4 = gB[mc], C4 = gC[mc], T4 = gT[mc];
    float4 S4 = gS[mc], S24 = gS2[mc];

    // B matrices: 4x16 layout — VGPR0 holds K=0 (lo half) / K=2 (hi half), VGPR1 K=1/K=3
    v2f b1; b1.x = hi ? A4.z : A4.x; b1.y = hi ? 0.0f : A4.y;   // seg
    v2f b2; b2.x = hi ? B4.z : B4.x; b2.y = hi ? 0.0f : B4.y;   // seg/sigma^2
    v2f b3; b3.x = hi ? C4.z : C4.x; b3.y = hi ? 0.0f : C4.y;   // p0/sigma^2
    v2f bp; bp.x = hi ? T4.z : T4.x; bp.y = hi ? 0.0f : T4.y;   // tgt/sigma

    v8f G1 = wmma4(a_ln, b1);   // pc . seg
    v8f G2 = wmma4(a_ln, b2);   // pc . seg / sigma^2
    v8f G3 = wmma4(a_ln, b3);   // pc . p0  / sigma^2
    v8f Cp = wmma4(a_pt, bp);   // (pc/sigma) . (tgt/sigma)

#pragma unroll
    for (int j = 0; j < 8; j++) {
      float tpar = fminf(fmaxf((G1[j] - S4.x) * S4.y, 0.0f), 1.0f);
      float u2 = row2[j] - 2.0f * G3[j] + S4.z;   // |(pc-p0)/sigma|^2 (+BIG if masked)
      float us = G2[j] - S4.w;                    // u . s
      float d2l = u2 - 2.0f * tpar * us + tpar * tpar * S24.x;
      minL[j] = fminf(minL[j], d2l);
      float d2p = row2[j] + S24.y - 2.0f * Cp[j]; // point distance (+BIG if masked)
      minP[j] = fminf(minP[j], d2p);
    }
  }

  // reduce min over the 16 columns held in each 16-lane half
#pragma unroll
  for (int j = 0; j < 8; j++) {
#pragma unroll
    for (int m = 8; m >= 1; m >>= 1) {
      minL[j] = fminf(minL[j], __shfl_xor(minL[j], m));
      minP[j] = fminf(minP[j], __shfl_xor(minP[j], m));
    }
  }

  float sumW = 0.0f, sumM = 0.0f;
  if (r == 0) {
#pragma unroll
    for (int j = 0; j < 8; j++) {
      int n = rowBase + hi * 8 + j;
      float pmn = pmask[b * NP + n] ? 1.0f : 0.0f;
      float ll = expf(-0.5f * minL[j]);       // max exp = exp(-0.5*min d2)
      float lp = expf(-0.5f * minP[j]);
      float lab = fmaxf(fminf(0.6f * ll + 0.4f * lp, 1.0f), 0.0f);
      sumW += (1.0f - lab) * pmn;
      sumM += pmn;
    }
  }
  sumW += __shfl_xor(sumW, 16);
  sumM += __shfl_xor(sumM, 16);
  if (lane == 0) {
    int pidx = b * 32 + tb * 8 + wave;  // 1024 per-wave partials
    pathW[pidx] = sumW;
    pathM[pidx] = sumM;
  }
}

// -------- Kernel 4: smoothness loss per batch -------------------------------
__global__ __launch_bounds__(256) void k_smooth(const float* __restrict__ pp,
                                                const int* __restrict__ pmask,
                                                float* __restrict__ smoothPart) {
  int b = blockIdx.x;
  int tid = threadIdx.x;
  const float* base = pp + (size_t)b * NP * 3;
  const int* mk = pmask + b * NP;
  float num = 0.0f, den = 0.0f;
  for (int i = tid; i < NP - 1; i += 256) {
    float dx = base[(i + 1) * 3 + 0] - base[i * 3 + 0];
    float dy = base[(i + 1) * 3 + 1] - base[i * 3 + 1];
    float dz = base[(i + 1) * 3 + 2] - base[i * 3 + 2];
    float m = (mk[i] && mk[i + 1]) ? 1.0f : 0.0f;
    num += (dx * dx + dy * dy + dz * dz) * m;
    den += m;
  }
  __shared__ float sn[256], sd[256];
  sn[tid] = num; sd[tid] = den;
  __syncthreads();
  for (int off = 128; off > 0; off >>= 1) {
    if (tid < off) { sn[tid] += sn[tid + off]; sd[tid] += sd[tid + off]; }
    __syncthreads();
  }
  if (tid == 0) smoothPart[b] = sn[0] / (sd[0] + EPSC);
}

// -------- Kernel 5: CAE MSE partial sums ------------------------------------
__global__ __launch_bounds__(256) void k_cae(const float* __restrict__ rec,
                                             const float* __restrict__ obs,
                                             float* __restrict__ caePart, int n) {
  int tid = blockIdx.x * 256 + threadIdx.x;
  float acc = 0.0f;
  for (int i = tid; i < n; i += gridDim.x * 256) {
    float d = rec[i] - obs[i];
    acc += d * d;
  }
  __shared__ float s[256];
  s[threadIdx.x] = acc;
  __syncthreads();
  for (int off = 128; off > 0; off >>= 1) {
    if (threadIdx.x < off) s[threadIdx.x] += s[threadIdx.x + off];
    __syncthreads();
  }
  if (threadIdx.x == 0) caePart[blockIdx.x] = s[0];
}

// -------- Kernel 6: deterministic final combine -----------------------------
__global__ __launch_bounds__(256) void k_final(const float* __restrict__ pathW,
                                               const float* __restrict__ pathM,
                                               const float* __restrict__ smoothPart,
                                               const float* __restrict__ caePart,
                                               float* __restrict__ out, float caeN) {
  int tid = threadIdx.x;
  float sw = 0.0f, sm = 0.0f, ss = 0.0f, sc = 0.0f;
  for (int i = tid; i < 1024; i += 256) { sw += pathW[i]; sm += pathM[i]; }
  if (tid < 32)  ss = smoothPart[tid];
  sc = caePart[tid];
  __shared__ float a0[256], a1[256], a2[256], a3[256];
  a0[tid] = sw; a1[tid] = sm; a2[tid] = ss; a3[tid] = sc;
  __syncthreads();
  for (int off = 128; off > 0; off >>= 1) {
    if (tid < off) {
      a0[tid] += a0[tid + off]; a1[tid] += a1[tid + off];
      a2[tid] += a2[tid + off]; a3[tid] += a3[tid + off];
    }
    __syncthreads();
  }
  if (tid == 0) {
    float loss_path   = a0[0] / a1[0];
    float loss_smooth = a2[0] / (float)NB;
    float loss_cae    = a3[0] / caeN;
    out[0] = 0.8f * loss_smooth + loss_path + 0.5f * loss_cae;
    out[1] = loss_path;
    out[2] = loss_smooth;
    out[3] = loss_cae;
  }
}

extern "C" void kernel_launch(void* const* d_in, const int* in_sizes, int n_in,
                              void* d_out, int out_size, void* d_ws, size_t ws_size,
                              hipStream_t stream) {
  const float* pp    = (const float*)d_in[0];
  const int*   pmask = (const int*)d_in[1];
  const float* tgt   = (const float*)d_in[2];
  const int*   tmask = (const int*)d_in[3];
  const float* rec   = (const float*)d_in[4];
  const float* obs   = (const float*)d_in[5];
  float* ws = (float*)d_ws;

  float4* segA    = (float4*)(ws + OFF_SEGA);
  float4* segB    = (float4*)(ws + OFF_SEGB);
  float4* segC    = (float4*)(ws + OFF_SEGC);
  float4* tgtS    = (float4*)(ws + OFF_TGTS);
  float4* scal    = (float4*)(ws + OFF_SCAL);
  float4* scal2   = (float4*)(ws + OFF_SCAL2);
  float4* sigInv  = (float4*)(ws + OFF_SIGI);
  float4* sigInv2 = (float4*)(ws + OFF_SIGI2);
  float*  pathW   = ws + OFF_PATHW;
  float*  pathM   = ws + OFF_PATHM;
  float*  smoothP = ws + OFF_SMOOTH;
  float*  caeP    = ws + OFF_CAE;

  int nCae = in_sizes[4];

  k_sigma<<<NB, 256, 0, stream>>>(pp, sigInv, sigInv2);
  k_prep<<<(NB * NG) / 256, 256, 0, stream>>>(tgt, tmask, sigInv, sigInv2,
                                              segA, segB, segC, tgtS, scal, scal2);
  k_main<<<NB * (NP / 128), 256, 0, stream>>>(pp, pmask, segA, segB, segC, tgtS,
                                              scal, scal2, sigInv, pathW, pathM);
  k_smooth<<<NB, 256, 0, stream>>>(pp, pmask, smoothP);
  k_cae<<<256, 256, 0, stream>>>(rec, obs, caeP, nCae);
  k_final<<<1, 256, 0, stream>>>(pathW, pathM, smoothP, caeP, (float*)d_out, (float)nCae);
}